// Rule_30528627540082
// MI455X (gfx1250) — compile-verified
//
#include <hip/hip_runtime.h>
#include <stdint.h>

// Ising-style cross-stencil update, B=16, H=W=1024, radius 2, wrap padding.
// HBM-bound (~290 MB @ 23.3 TB/s). Strategy:
//   - interior row-strips: one TDM tensor_load_to_lds DMA of the 12x1024 f32
//     spin tile (2D descriptor, TENSORcnt-tracked)
//   - wrap-around strips: per-lane global_load_async_to_lds_b128 (ASYNCcnt)
//   - all other IO: float4 (b128) loads/stores

typedef __attribute__((ext_vector_type(4))) uint32_t u32x4;
typedef __attribute__((ext_vector_type(8))) uint32_t u32x8;

namespace {
constexpr int Hh = 1024;
constexpr int Ww = 1024;
constexpr int HWc = Hh * Ww;
constexpr int TR = 8;                      // rows computed per block
constexpr int HALO = 2;                    // stencil radius
constexpr int LROWS = TR + 2 * HALO;       // 12 LDS rows
constexpr int THREADS = 256;               // 8 wave32
constexpr int CPR = Ww / 4;                // float4 chunks per row = 256
constexpr float HK_MAXf = 8.0f;
constexpr float DROPf = 0.75f;
}

__global__ __launch_bounds__(THREADS)
void ising_cross_kernel(const float* __restrict__ x,
                        const float* __restrict__ rand_u,
                        float* __restrict__ out)
{
    // 12 rows x 1024 floats = 48 KB spin tile
    __shared__ float4 smem[LROWS * CPR];

    const int tid   = threadIdx.x;
    const int tiles = Hh / TR;                       // 128 strips per image
    const int b     = blockIdx.x / tiles;
    const int y0    = (blockIdx.x - b * tiles) * TR;

    const float* spins = x + (size_t)b * 2 * HWc;    // channel 0
    const float* mom   = spins + HWc;                // channel 1
    float* outs = out + (size_t)b * 2 * HWc;
    float* outm = outs + HWc;

    const uint32_t lds0 = (uint32_t)(uintptr_t)(&smem[0]);
    const int gstart = y0 - HALO;
    const bool interior = (gstart >= 0) && (gstart + LROWS <= Hh);

    if (interior) {
        // ---- TDM path: single 2D tensor DMA, issued by wave 0 only ----
        // (TDM ignores EXEC, so guard whole waves to avoid duplicate DMAs.)
        if (tid < 32) {
            const uint64_t ga = (uint64_t)(uintptr_t)spins
                              + (uint64_t)gstart * (uint64_t)(Ww * 4);
            u32x4 g0;
            g0[0] = 1u;                                     // count=1, user D#
            g0[1] = lds0;                                   // lds_addr (bytes)
            g0[2] = (uint32_t)ga;                           // global_addr[31:0]
            g0[3] = ((uint32_t)(ga >> 32) & 0x01FFFFFFu)    // global_addr[56:32]
                  | 0x80000000u;                            // type = 2 ("image")
            u32x8 g1;
            g1[0] = 0x00020000u;                  // wg_mask=0, data_size=2 (4B)
            g1[1] = (uint32_t)Ww << 16;           // tensor_dim0[15:0] = 1024
            g1[2] = (uint32_t)LROWS << 16;        // dim0 hi=0 | tensor_dim1 lo=12
            g1[3] = (uint32_t)Ww << 16;           // dim1 hi=0 | tile_dim0 = 1024
            g1[4] = (uint32_t)LROWS;              // tile_dim1=12, tile_dim2=0
            g1[5] = (uint32_t)Ww;                 // tensor_dim0_stride lo = 1024
            g1[6] = 0u;                           // stride hi | dim1_stride lo
            g1[7] = 0u;                           // dim1_stride hi (unused, 2D)
            asm volatile("tensor_load_to_lds %0, %1"
                         :: "s"(g0), "s"(g1)
                         : "memory");
            __builtin_amdgcn_s_wait_tensorcnt(0);
        }
    } else {
        // ---- Wrap-around strips: per-lane async b128 global->LDS ----
        const uint64_t sbase = (uint64_t)(uintptr_t)spins;
        #pragma unroll
        for (int i = 0; i < (LROWS * CPR) / THREADS; ++i) {   // 12 iterations
            const int c   = tid + i * THREADS;                // 16B chunk index
            const int lr  = c >> 6;                           // LDS row
            const int xc  = c & 63;
            const int row = (y0 - HALO + lr) & (Hh - 1);      // vertical wrap
            const uint32_t voff = (uint32_t)(row * (Ww * 4) + xc * 16);
            const uint32_t la   = lds0 + (uint32_t)c * 16u;
            asm volatile("global_load_async_to_lds_b128 %0, %1, %2 offset:0"
                         :: "v"(la), "v"(voff), "s"(sbase)
                         : "memory");
        }
        asm volatile("s_wait_asynccnt 0x0" ::: "memory");
    }
    __syncthreads();

    // ---- Compute: 8 rows, 4 pixels per thread per row, all float4 ----
    #pragma unroll
    for (int r = 0; r < TR; ++r) {
        const int y = y0 + r;
        const float4 cc  = smem[(r + 2) * CPR + tid];
        const float4 lf  = smem[(r + 2) * CPR + ((tid - 1) & (CPR - 1))];
        const float4 rt  = smem[(r + 2) * CPR + ((tid + 1) & (CPR - 1))];
        const float4 um2 = smem[(r + 0) * CPR + tid];
        const float4 um1 = smem[(r + 1) * CPR + tid];
        const float4 dp1 = smem[(r + 3) * CPR + tid];
        const float4 dp2 = smem[(r + 4) * CPR + tid];

        const size_t roff = (size_t)y * Ww + (size_t)tid * 4;
        const float4 m4 = *reinterpret_cast<const float4*>(mom + roff);
        const float4 u4 = *reinterpret_cast<const float4*>(rand_u + roff);

        const float w[12] = { lf.x, lf.y, lf.z, lf.w,
                              cc.x, cc.y, cc.z, cc.w,
                              rt.x, rt.y, rt.z, rt.w };
        const float vs[4] = { um2.x + um1.x + dp1.x + dp2.x,
                              um2.y + um1.y + dp1.y + dp2.y,
                              um2.z + um1.z + dp1.z + dp2.z,
                              um2.w + um1.w + dp1.w + dp2.w };
        const float mm[4] = { m4.x, m4.y, m4.z, m4.w };
        const float uu[4] = { u4.x, u4.y, u4.z, u4.w };

        float so[4], mo[4];
        #pragma unroll
        for (int j = 0; j < 4; ++j) {
            const int k = 4 + j;
            const float Js = (w[k - 2] + w[k - 1] + w[k + 1] + w[k + 2]) + vs[j];
            const float s  = w[k];
            const float Hk = mm[j] - 2.0f * s * Js;       // momentum - delta_e
            const bool flip = (Hk >= 0.0f) && (Hk <= HK_MAXf) && (uu[j] > DROPf);
            so[j] = flip ? -s : s;
            mo[j] = flip ? Hk : mm[j];
        }
        *reinterpret_cast<float4*>(outs + roff) = make_float4(so[0], so[1], so[2], so[3]);
        *reinterpret_cast<float4*>(outm + roff) = make_float4(mo[0], mo[1], mo[2], mo[3]);
    }
}

extern "C" void kernel_launch(void* const* d_in, const int* in_sizes, int n_in,
                              void* d_out, int out_size, void* d_ws, size_t ws_size,
                              hipStream_t stream) {
    (void)in_sizes; (void)n_in; (void)out_size; (void)d_ws; (void)ws_size;
    const float* x      = (const float*)d_in[0];   // (16, 2, 1024, 1024) f32
    const float* rand_u = (const float*)d_in[1];   // (1024, 1024) f32
    float* out          = (float*)d_out;           // (16, 2, 1024, 1024) f32

    const int B = 16;
    dim3 grid(B * (Hh / TR));   // 2048 blocks
    dim3 block(THREADS);
    ising_cross_kernel<<<grid, block, 0, stream>>>(x, rand_u, out);
}